// SinkhornMatcher_20770461843949
// MI455X (gfx1250) — compile-verified
//
#include <hip/hip_runtime.h>
#include <cstdint>
#include <cstddef>

#define BATCH 8
#define NN    512
#define DD    256
#define MM    32
#define MAX_ITERS 100
#define EPS   1e-4f

typedef float v2f __attribute__((ext_vector_type(2)));
typedef float v8f __attribute__((ext_vector_type(8)));
typedef unsigned int v4u __attribute__((ext_vector_type(4)));
typedef int v4i __attribute__((ext_vector_type(4)));
typedef int v8i __attribute__((ext_vector_type(8)));

// ---------------------------------------------------------------------------
// Phase 1: projections, templated on side so each instantiation is a clean
// straight-line V_WMMA_F32_16X16X4_F32 K-loop (no runtime-proj branches).
//   PROJ=0: a = (outEmb + pos) @ W_out + b_out
//   PROJ=1: b = (mask ? pad : inEmb) @ W_in + b_in
// A frag (16x4 f32): lane l holds row (l&15), K = kbase + (l>>4)*2 + {0,1}.
// B frag (4x16 f32): lane l holds col (l&15), same K pair.
// C/D (16x16 f32): vgpr r -> row r + 8*(l>>4), col (l&15).
// ---------------------------------------------------------------------------
template <int PROJ>
__global__ __launch_bounds__(64) void proj_kernel(
    const float* __restrict__ E, const unsigned char* __restrict__ mask,
    const float* __restrict__ pad, const float* __restrict__ pos,
    const float* __restrict__ W, const float* __restrict__ bias,
    float* __restrict__ dst)
{
    const int tile = blockIdx.x;        // 0..(B*N/16 - 1)
    const int lane = threadIdx.x & 31;
    const int wave = threadIdx.x >> 5;  // column half (M=32 -> two 16x16 tiles)
    const int c    = wave * 16 + (lane & 15);
    const int rowA = tile * 16 + (lane & 15);
    const int nIdx = rowA & (NN - 1);
    const int klo  = (lane >> 4) * 2;

    const float* Erow = E + (size_t)rowA * DD;
    const float* Prow = pos + (size_t)nIdx * DD;
    const float* Wc   = W + c;
    bool msk = false;
    if (PROJ == 1) msk = (mask[rowA] != 0);

    v8f acc = {};
#pragma unroll 4
    for (int k = 0; k < DD; k += 4) {
        const int d0 = k + klo;
        v2f aF, bF;
        if (PROJ == 1) {
            aF.x = msk ? pad[d0]     : Erow[d0];
            aF.y = msk ? pad[d0 + 1] : Erow[d0 + 1];
        } else {
            aF.x = Erow[d0]     + Prow[d0];
            aF.y = Erow[d0 + 1] + Prow[d0 + 1];
        }
        bF.x = Wc[(size_t)d0 * MM];
        bF.y = Wc[(size_t)(d0 + 1) * MM];
        acc = __builtin_amdgcn_wmma_f32_16x16x4_f32(
            false, aF, false, bF, (short)0, acc, false, false);
    }

    const float bv = bias[c];
    const int rbase = tile * 16 + ((lane >> 4) ? 8 : 0);
#pragma unroll
    for (int r = 0; r < 8; ++r)
        dst[(size_t)(rbase + r) * MM + c] = acc[r] + bv;
}

// ---------------------------------------------------------------------------
// TDM: DMA one 64x32 f32 tile (contiguous 8 KB) global -> LDS.
// D# per CDNA5 ISA. EXEC ignored by tensor ops.
// ---------------------------------------------------------------------------
__device__ inline void tdm_load_tile(unsigned lds_off, const float* gptr)
{
    const unsigned long long ga = (unsigned long long)(uintptr_t)gptr;
    v4u g0;
    g0.x = 1u;                                         // count=1, user mode
    g0.y = lds_off;                                    // lds_addr (bytes)
    g0.z = (unsigned)(ga & 0xffffffffull);             // global_addr[31:0]
    g0.w = (unsigned)((ga >> 32) & 0x01ffffffull)      // global_addr[56:32]
         | (2u << 30);                                 // type=2 ("image")
    v8i g1;
    g1[0] = (int)(2u << 16);       // data_size = 2 -> 4 bytes; no multicast
    g1[1] = (int)(32u << 16);      // tensor_dim0 = 32         (bits 63:48)
    g1[2] = (int)(512u << 16);     // tensor_dim1 = 512        (bits 95:80)
    g1[3] = (int)(32u << 16);      // tile_dim0 = 32           (bits 127:112)
    g1[4] = 64;                    // tile_dim1 = 64           (bits 143:128)
    g1[5] = 32;                    // tensor_dim0_stride = 32  (bits 191:160)
    g1[6] = 0;
    g1[7] = 0;
    v4i gz = {0, 0, 0, 0};
#if defined(__clang_major__) && __clang_major__ >= 23
    v8i gz8 = {};
    __builtin_amdgcn_tensor_load_to_lds(g0, g1, gz, gz, gz8, 0);
#else
    __builtin_amdgcn_tensor_load_to_lds(g0, g1, gz, gz, 0);
#endif
}

// ---------------------------------------------------------------------------
// Phase 2: 64x64 cost tiles; TDM-staged LDS tiles; per-thread 4x4 sub-block;
// per-batch sum via wave reduce + global_atomic_add_f32.
// ---------------------------------------------------------------------------
__global__ __launch_bounds__(256) void cost_kernel(
    const float* __restrict__ aP, const float* __restrict__ bP,
    float* __restrict__ C, float* __restrict__ sums)
{
    __shared__ __align__(16) float shA[64 * MM];
    __shared__ __align__(16) float shB[64 * MM];
    const int b  = blockIdx.z;
    const int i0 = blockIdx.y * 64;
    const int j0 = blockIdx.x * 64;
    const int tid = threadIdx.x;

    if (tid < 32) {  // wave 0 drives both DMAs (tensor ops ignore EXEC)
        tdm_load_tile((unsigned)(uintptr_t)&shA[0], aP + ((size_t)b * NN + i0) * MM);
        tdm_load_tile((unsigned)(uintptr_t)&shB[0], bP + ((size_t)b * NN + j0) * MM);
        __builtin_amdgcn_s_wait_tensorcnt(0);
    }
    __syncthreads();

    const int ti = tid >> 4, tj = tid & 15;
    float total = 0.0f;
    float* Cb = C + (size_t)b * NN * NN;
#pragma unroll
    for (int p = 0; p < 4; ++p) {
        const float* ar = &shA[(ti * 4 + p) * MM];
        float tmp[4];
#pragma unroll
        for (int q = 0; q < 4; ++q) {
            const float* br = &shB[(tj * 4 + q) * MM];
            float s = 0.0f;
#pragma unroll
            for (int m = 0; m < MM; ++m) s += fabsf(ar[m] - br[m]);
            tmp[q] = s;
            total += s;
        }
        float4 res = make_float4(tmp[0], tmp[1], tmp[2], tmp[3]);
        *(float4*)&Cb[(size_t)(i0 + ti * 4 + p) * NN + j0 + tj * 4] = res;
    }
    for (int off = 16; off; off >>= 1) total += __shfl_xor(total, off, 32);
    if ((tid & 31) == 0) atomicAdd(&sums[b], total);
}

__global__ void init_sums(float* sums)
{
    if (threadIdx.x < BATCH) sums[threadIdx.x] = 0.0f;
}

// ---------------------------------------------------------------------------
// Phase 3: Sinkhorn, one workgroup per batch (512 thr = 16 wave32).
// Online logsumexp with ONE v_exp_f32 per element:
//   e = exp(-|x-m|);  s = (x>m) ? s*e+1 : s+e;  m = max(m,x)
// Row pass: wave-per-row, float4 global reads (b128) + float4 LDS reads.
// Col pass: column-per-thread, lane-consecutive -> coalesced.
// ---------------------------------------------------------------------------
__global__ __launch_bounds__(512) void sinkhorn_kernel(
    const float* __restrict__ C, const float* __restrict__ sums,
    float* __restrict__ out)
{
    __shared__ __align__(16) float u[NN];
    __shared__ __align__(16) float v[NN];
    const int b    = blockIdx.x;
    const int tid  = threadIdx.x;
    const int lane = tid & 31;
    const int wave = tid >> 5;
    const float inv  = 1.0f / (EPS * sums[b]);
    const float logm = -__logf((float)NN);
    const float* Cb  = C + (size_t)b * NN * NN;

    u[tid] = 0.0f;
    v[tid] = 0.0f;
    __syncthreads();

    for (int it = 0; it < MAX_ITERS; ++it) {
        // ---- row pass: u[i] = logm - lse_j(v[j] - C[i,j]*inv) ----
        for (int r = 0; r < 32; ++r) {
            const int i = wave * 32 + r;
            const float4* rowv = (const float4*)(Cb + (size_t)i * NN);
            const float4* vv   = (const float4*)v;
            float m = -3.0e38f, s = 0.0f;
#pragma unroll
            for (int t = 0; t < 4; ++t) {
                const int idx = t * 32 + lane;       // 128 float4 = 512 floats
                float4 cq = rowv[idx];
                float4 vq = vv[idx];
                float xs[4] = { vq.x - cq.x * inv, vq.y - cq.y * inv,
                                vq.z - cq.z * inv, vq.w - cq.w * inv };
#pragma unroll
                for (int q = 0; q < 4; ++q) {
                    float d = xs[q] - m;
                    float e = __expf(-fabsf(d));
                    bool gt = d > 0.0f;
                    s = gt ? s * e + 1.0f : s + e;
                    m = gt ? xs[q] : m;
                }
            }
            for (int off = 16; off; off >>= 1) {
                float mo = __shfl_xor(m, off, 32);
                float so = __shfl_xor(s, off, 32);
                float d  = mo - m;
                float e  = __expf(-fabsf(d));
                bool gt  = d > 0.0f;
                s = gt ? s * e + so : s + so * e;
                m = gt ? mo : m;
            }
            if (lane == 0) u[i] = logm - (m + __logf(s));
        }
        __syncthreads();
        // ---- col pass: v[j] = logm - lse_i(u[i] - C[i,j]*inv) ----
        {
            const float* col = Cb + tid;
            float m = -3.0e38f, s = 0.0f;
#pragma unroll 4
            for (int i = 0; i < NN; ++i) {
                float x = u[i] - col[(size_t)i * NN] * inv;
                float d = x - m;
                float e = __expf(-fabsf(d));
                bool gt = d > 0.0f;
                s = gt ? s * e + 1.0f : s + e;
                m = gt ? x : m;
            }
            v[tid] = logm - (m + __logf(s));
        }
        __syncthreads();
    }

    const float vt = v[tid];
    float* Ob = out + (size_t)b * NN * NN;
#pragma unroll 4
    for (int i = 0; i < NN; ++i)
        Ob[(size_t)i * NN + tid] =
            __expf(u[i] + vt - Cb[(size_t)i * NN + tid] * inv);
}

// ---------------------------------------------------------------------------
extern "C" void kernel_launch(void* const* d_in, const int* in_sizes, int n_in,
                              void* d_out, int out_size, void* d_ws, size_t ws_size,
                              hipStream_t stream)
{
    (void)in_sizes; (void)n_in; (void)out_size; (void)ws_size;
    const float* inEmb        = (const float*)d_in[0];
    const unsigned char* mask = (const unsigned char*)d_in[1];
    const float* outEmb       = (const float*)d_in[2];
    const float* pad          = (const float*)d_in[3];
    const float* pos          = (const float*)d_in[4];
    const float* Win          = (const float*)d_in[5];
    const float* bin          = (const float*)d_in[6];
    const float* Wout         = (const float*)d_in[7];
    const float* bout         = (const float*)d_in[8];
    float* out = (float*)d_out;

    // workspace (floats): aP[8*512*32] | bP[8*512*32] | C[8*512*512] | sums[8]
    float* w    = (float*)d_ws;
    float* aP   = w;
    float* bP   = aP + (size_t)BATCH * NN * MM;
    float* C    = bP + (size_t)BATCH * NN * MM;
    float* sums = C  + (size_t)BATCH * NN * NN;

    init_sums<<<1, 32, 0, stream>>>(sums);
    proj_kernel<0><<<BATCH * NN / 16, 64, 0, stream>>>(
        outEmb, mask, pad, pos, Wout, bout, aP);
    proj_kernel<1><<<BATCH * NN / 16, 64, 0, stream>>>(
        inEmb, mask, pad, pos, Win, bin, bP);
    cost_kernel<<<dim3(NN / 64, NN / 64, BATCH), 256, 0, stream>>>(aP, bP, C, sums);
    sinkhorn_kernel<<<BATCH, NN, 0, stream>>>(C, sums, out);
}